// Linear_66477503808155
// MI455X (gfx1250) — compile-verified
//
#include <hip/hip_runtime.h>
#include <stdint.h>

typedef float v2f __attribute__((ext_vector_type(2)));
typedef float v8f __attribute__((ext_vector_type(8)));

#define AS1 __attribute__((address_space(1)))
#define AS3 __attribute__((address_space(3)))

static constexpr int kMul     = 512;   // u and w multiplicity
static constexpr int kRow     = 4608;  // floats per z-row of x1/out
static constexpr int kZB      = 16;    // z-rows per block (one 16-row M tile)
static constexpr int kThreads = 256;   // 8 waves (wave32)
static constexpr int KC       = 64;    // K-panel depth (u per panel)
static constexpr int NP       = kMul / KC;  // 8 panels
static constexpr int kUPad    = KC + 4;     // 68: bank-conflict-free LDS row stride

#if defined(__has_builtin)
#if __has_builtin(__builtin_amdgcn_global_load_async_to_lds_b32) && \
    __has_builtin(__builtin_amdgcn_s_wait_asynccnt)
#define USE_ASYNC 1
#else
#define USE_ASYNC 0
#endif
#else
#define USE_ASYNC 0
#endif

// Out[z, w, i] = c * x2[z] * sum_u X[z, u, i] * W[u, w]   (per l; D = 2l+1)
//
// V_WMMA_F32_16X16X4_F32 mapping (D = A(16x4)*B(4x16) + C):
//   A[m,k] = X[z0+m, u=k, i]  (from double-buffered LDS K-panels)
//   B[k,n] = W[u=k, wb+n]     (L2-resident weight stream)
//   C/D    = Out tile, rows z0+m, cols wb+n
// K is pipelined in 8 panels of 64: async-copy panel p+1 into LDS buffer
// (p+1)&1 while WMMA-ing panel p from buffer p&1.
template <int D>
__global__ __launch_bounds__(kThreads)
void e3nn_linear_wmma(const float* __restrict__ x1, const float* __restrict__ x2,
                      const float* __restrict__ W, float* __restrict__ out,
                      int off, float cscale) {
  __shared__ float xs[2][D * kZB * kUPad];  // <= 43.5 KB for D=5

  const int tid = threadIdx.x;
  const int z0  = blockIdx.x * kZB;

  // Stage one K-panel into LDS buffer b, transposing (u,i) -> [i][z][u_local].
  // Global side: consecutive threads touch consecutive dwords of the
  // contiguous KC*D-float span of each z-row (coalesced). The transpose is
  // folded into the per-lane LDS destination address of the async copy.
  auto stage = [&](int p, int b) {
    const int gbase = off + p * (KC * D);
#pragma unroll
    for (int t = 0; t < 4 * D; ++t) {          // 16*KC*D dwords / 256 threads
      const int idx = tid + t * kThreads;
      const int z   = idx / (KC * D);
      const int r   = idx - z * (KC * D);
      const int u   = r / D;
      const int i   = r - u * D;
      const float* gp = x1 + (size_t)(z0 + z) * kRow + gbase + r;
      float* lp = &xs[b][(i * kZB + z) * kUPad + u];
#if USE_ASYNC
      __builtin_amdgcn_global_load_async_to_lds_b32(
          (AS1 int*)gp, (AS3 int*)lp, 0, 0);
#else
      *lp = __builtin_nontemporal_load(gp);
#endif
    }
  };

  const int lane = tid & 31;
  const int wave = tid >> 5;
  const int m    = lane & 15;  // M row / N col within 16x16 tile
  const int hk   = lane >> 4;  // lane half -> K pair {0,1} vs {2,3}
  const int wb   = wave * 64;  // this wave's 64-wide w stripe (8*64 = 512)

  v8f acc[D][4] = {};  // D i-slices x 4 w-tiles (<=160 VGPRs for D=5)

  stage(0, 0);
  for (int p = 0; p < NP; ++p) {
    const int b = p & 1;
#if USE_ASYNC
    __builtin_amdgcn_s_wait_asynccnt(0);  // my panel-p writes have landed
#endif
    __syncthreads();                       // everyone's writes visible
    if (p + 1 < NP) stage(p + 1, b ^ 1);   // overlap copy-in with compute

#pragma unroll 4
    for (int ks = 0; ks < KC / 4; ++ks) {
      const int kl = 4 * ks + 2 * hk;      // K offset inside panel (lane half)
      v2f a[D];
#pragma unroll
      for (int i = 0; i < D; ++i)          // conflict-free ds_load_b64
        a[i] = *(const v2f*)&xs[b][(i * kZB + m) * kUPad + kl];
#pragma unroll
      for (int wt = 0; wt < 4; ++wt) {
        const float* wp = W + (size_t)(p * KC + kl) * kMul + (wb + wt * 16 + m);
        v2f bb = {wp[0], wp[kMul]};        // W rows k, k+1 (L2-resident)
#pragma unroll
        for (int i = 0; i < D; ++i)
          acc[i][wt] = __builtin_amdgcn_wmma_f32_16x16x4_f32(
              false, a[i], false, bb, (short)0, acc[i][wt], false, false);
      }
    }
    __syncthreads();  // all reads of buffer b done before it is re-staged
  }

  // Scale by c*x2[z]; each lane writes its D consecutive floats as one packed
  // store -> 16 lanes cover 16*D contiguous floats per (v, wt).
#pragma unroll
  for (int v = 0; v < 8; ++v) {
    const int z   = z0 + v + 8 * hk;
    const float s = cscale * x2[z];
    float* orow   = out + (size_t)z * kRow + off;
#pragma unroll
    for (int wt = 0; wt < 4; ++wt) {
      const int w = wb + wt * 16 + m;
      float tmp[D];
#pragma unroll
      for (int i = 0; i < D; ++i) tmp[i] = acc[i][wt][v] * s;
      __builtin_memcpy(orow + (size_t)w * D, tmp, D * sizeof(float));
    }
  }
}

extern "C" void kernel_launch(void* const* d_in, const int* in_sizes, int n_in,
                              void* d_out, int out_size, void* d_ws, size_t ws_size,
                              hipStream_t stream) {
  const float* x1 = (const float*)d_in[0];
  const float* x2 = (const float*)d_in[1];  // 1x0e input, folded as per-z scale
  const float* W0 = (const float*)d_in[2];
  const float* W1 = (const float*)d_in[3];
  const float* W2 = (const float*)d_in[4];
  float* out = (float*)d_out;

  const int N = in_sizes[0] / kRow;      // 16384
  const float c = 0.04419417382415922f;  // 1/sqrt(512)
  dim3 grid(N / kZB), block(kThreads);

  e3nn_linear_wmma<1><<<grid, block, 0, stream>>>(x1, x2, W0, out, 0, c);
  e3nn_linear_wmma<3><<<grid, block, 0, stream>>>(x1, x2, W1, out, 512, c);
  e3nn_linear_wmma<5><<<grid, block, 0, stream>>>(x1, x2, W2, out, 2048, c);
}